// GNN_53300544143387
// MI455X (gfx1250) — compile-verified
//
#include <hip/hip_runtime.h>
#include <math.h>

typedef __attribute__((ext_vector_type(2))) float v2f;
typedef __attribute__((ext_vector_type(8))) float v8f;

#define F_IN  128
#define F_HID 16

// -------- hardware f32 atomic add (global_atomic_add_f32, no CAS loop) -----
__device__ __forceinline__ void atom_add_f32(float* p, float v) {
    unsafeAtomicAdd(p, v);
}

// -------- zero accumulator workspace ---------------------------------------
__global__ void gnn_zero_f32(float* __restrict__ p, int n) {
    int i = blockIdx.x * blockDim.x + threadIdx.x;
    int stride = gridDim.x * blockDim.x;
    for (; i < n; i += stride) p[i] = 0.0f;
}

// -------- z0 = x @ W0 via v_wmma_f32_16x16x4_f32 ---------------------------
// One wave per 16-row tile; K=128 swept as 32 chunks of 4.
__global__ __launch_bounds__(256) void gnn_gemm_wmma(
    const float* __restrict__ x, const float* __restrict__ W0,
    float* __restrict__ z0, int n_nodes, int n_tiles) {
    const int lane = threadIdx.x & 31;
    const int wave = threadIdx.x >> 5;
    const int tile = blockIdx.x * 8 + wave;   // 8 waves / 256-thread block
    if (tile >= n_tiles) return;

    const int half = lane >> 4;               // 0: lanes 0-15, 1: lanes 16-31
    const int l    = lane & 15;

    int row = tile * 16 + l;                  // A-matrix M index for this lane
    if (row >= n_nodes) row = n_nodes - 1;    // tail clamp (harmless reload)
    const float* xrow = x + (size_t)row * F_IN;

    // Preload B fragments (W0 is tile-invariant): per chunk c, lane holds
    // rows K = 4c+2*half, 4c+2*half+1 at column N = l.
    v2f bfrag[F_IN / 4];
#pragma unroll
    for (int c = 0; c < F_IN / 4; ++c) {
        const int k = c * 4 + 2 * half;
        v2f b;
        b.x = W0[(k + 0) * F_HID + l];
        b.y = W0[(k + 1) * F_HID + l];
        bfrag[c] = b;
    }

    v8f acc = {};
#pragma unroll
    for (int c = 0; c < F_IN / 4; ++c) {
        // A fragment: lane holds (K = 4c+2*half, 4c+2*half+1) of row M.
        v2f a = *(const v2f*)(xrow + c * 4 + 2 * half);
        acc = __builtin_amdgcn_wmma_f32_16x16x4_f32(
            /*neg_a=*/false, a, /*neg_b=*/false, bfrag[c],
            /*c_mod=*/(short)0, acc, /*reuse_a=*/false, /*reuse_b=*/false);
    }

    // D layout: VGPR r -> M = r + 8*half, N = l.
#pragma unroll
    for (int r = 0; r < 8; ++r) {
        int m = tile * 16 + r + 8 * half;
        if (m < n_nodes) z0[(size_t)m * F_HID + l] = acc[r];
    }
}

// -------- SpMM layer 0: hpre[dst] += z0[src] * w  (16 lanes per edge) ------
__global__ void gnn_spmm16(const int* __restrict__ src, const int* __restrict__ dst,
                           const float* __restrict__ ew, const float* __restrict__ z0,
                           float* __restrict__ hpre, int n_edges) {
    long long t = (long long)blockIdx.x * blockDim.x + threadIdx.x;
    int e = (int)(t >> 4);
    int f = (int)(t & 15);
    if (e >= n_edges) return;
    int s = src[e];
    int d = dst[e];
    float v = z0[(size_t)s * F_HID + f] * ew[e];
    atom_add_f32(&hpre[(size_t)d * F_HID + f], v);
}

// -------- h = relu(hpre + b0);  z1 = h @ W1 --------------------------------
__global__ void gnn_relu_w1(const float* __restrict__ hpre, const float* __restrict__ b0,
                            const float* __restrict__ W1, float* __restrict__ z1, int n) {
    int i = blockIdx.x * blockDim.x + threadIdx.x;
    if (i >= n) return;
    const float4* hp = (const float4*)(hpre + (size_t)i * F_HID);
    float acc = 0.0f;
#pragma unroll
    for (int q = 0; q < 4; ++q) {
        float4 h4 = hp[q];
        float h0 = h4.x + b0[4 * q + 0]; h0 = h0 > 0.f ? h0 : 0.f;
        float h1 = h4.y + b0[4 * q + 1]; h1 = h1 > 0.f ? h1 : 0.f;
        float h2 = h4.z + b0[4 * q + 2]; h2 = h2 > 0.f ? h2 : 0.f;
        float h3 = h4.w + b0[4 * q + 3]; h3 = h3 > 0.f ? h3 : 0.f;
        acc += h0 * W1[4 * q + 0] + h1 * W1[4 * q + 1]
             + h2 * W1[4 * q + 2] + h3 * W1[4 * q + 3];
    }
    z1[i] = acc;
}

// -------- SpMM layer 1: outpre[dst] += z1[src] * w -------------------------
__global__ void gnn_spmm1(const int* __restrict__ src, const int* __restrict__ dst,
                          const float* __restrict__ ew, const float* __restrict__ z1,
                          float* __restrict__ outpre, int n_edges) {
    int e = blockIdx.x * blockDim.x + threadIdx.x;
    if (e >= n_edges) return;
    atom_add_f32(&outpre[dst[e]], z1[src[e]] * ew[e]);
}

// -------- out = sigmoid(outpre + b1) ---------------------------------------
__global__ void gnn_sigmoid(const float* __restrict__ outpre, const float* __restrict__ b1,
                            float* __restrict__ out, int n) {
    int i = blockIdx.x * blockDim.x + threadIdx.x;
    if (i >= n) return;
    float t = outpre[i] + b1[0];
    out[i] = 1.0f / (1.0f + __expf(-t));
}

extern "C" void kernel_launch(void* const* d_in, const int* in_sizes, int n_in,
                              void* d_out, int out_size, void* d_ws, size_t ws_size,
                              hipStream_t stream) {
    const float* x   = (const float*)d_in[0];
    const int*   src = (const int*)  d_in[1];
    const int*   dst = (const int*)  d_in[2];
    const float* ew  = (const float*)d_in[3];
    const float* W0  = (const float*)d_in[4];
    const float* b0  = (const float*)d_in[5];
    const float* W1  = (const float*)d_in[6];
    const float* b1  = (const float*)d_in[7];
    float* out = (float*)d_out;

    const int n_nodes = in_sizes[0] / F_IN;
    const int n_edges = in_sizes[1];

    // Workspace layout
    float* z0     = (float*)d_ws;
    float* hpre   = z0   + (size_t)n_nodes * F_HID;
    float* z1     = hpre + (size_t)n_nodes * F_HID;
    float* outpre = z1   + (size_t)n_nodes;

    // 1) zero the accumulators (hpre .. outpre region, z1 overwritten anyway)
    const int n_zero = n_nodes * (F_HID + 2);
    gnn_zero_f32<<<(n_zero + 255) / 256, 256, 0, stream>>>(hpre, n_zero);

    // 2) z0 = x @ W0 (WMMA)
    const int n_tiles = (n_nodes + 15) / 16;
    gnn_gemm_wmma<<<(n_tiles + 7) / 8, 256, 0, stream>>>(x, W0, z0, n_nodes, n_tiles);

    // 3) hpre[dst] += z0[src] * w
    const long long t1 = (long long)n_edges * F_HID;
    gnn_spmm16<<<(unsigned)((t1 + 255) / 256), 256, 0, stream>>>(src, dst, ew, z0, hpre, n_edges);

    // 4) z1 = relu(hpre + b0) @ W1
    gnn_relu_w1<<<(n_nodes + 255) / 256, 256, 0, stream>>>(hpre, b0, W1, z1, n_nodes);

    // 5) outpre[dst] += z1[src] * w
    gnn_spmm1<<<(n_edges + 255) / 256, 256, 0, stream>>>(src, dst, ew, z1, outpre, n_edges);

    // 6) out = sigmoid(outpre + b1)
    gnn_sigmoid<<<(n_nodes + 255) / 256, 256, 0, stream>>>(outpre, b1, out, n_nodes);
}